// MaskAttentionOp_90159953477782
// MI455X (gfx1250) — compile-verified
//
#include <hip/hip_runtime.h>

// Fused windowed multi-head attention + 1x1-conv projection for MI455X (gfx1250).
// fp32 end-to-end using V_WMMA_F32_16X16X4_F32 (kernel is HBM-bandwidth bound:
// ~0.83 GB mandatory traffic -> ~36us floor @ 23.3 TB/s; f32 WMMA compute fits
// under that, so no precision downgrade is warranted).
// Stage A: GLOBAL_LOAD_ASYNC_TO_LDS_B128 (ASYNCcnt), q/k/v never touch VGPRs.
// Stage B: fragment caching + full B-fragment preload -> 16-WMMA bursts.
// Stage C: row softmax, rpe+mask added via coalesced b128 (mask non-temporal).
// Stage D: V/P fragment preload, b128 X_T stores, 1/sum folded at store.
// Stage E: X_T layout makes B-fragments single ds_load_b64; W b64 from L2.

typedef __attribute__((ext_vector_type(2))) float v2f;
typedef __attribute__((ext_vector_type(4))) float v4f;
typedef __attribute__((ext_vector_type(8))) float v8f;
typedef __attribute__((address_space(3))) float lds_f;

#define NWIN 4096   // windows (Bn)
#define CCH  192    // channels
#define NN   64     // tokens per window (8x8)
#define NHD  6      // heads
#define DH   32     // head dim
#define SROW 66     // padded score-row stride: even (b64-mergeable), conflict-free
#define XST  196    // X_T row stride (floats): 16B-aligned rows, bank stride 4

// LDS float offsets
#define LDS_Q 0               // raw q; X_T[64][XST] overwrites [0,12544) later
#define LDS_K 12288
#define LDS_V 24576
#define LDS_S 36864           // 6 * 64 * SROW = 25344
#define LDS_I (LDS_S + NHD * NN * SROW)        // 384 x (1/rowsum)
#define LDS_FLOATS (LDS_I + NHD * NN)          // 62592 floats = 250368 B

__device__ __forceinline__ v8f wmma_f32(v2f a, v2f b, v8f c) {
    // 8 args: (neg_a, A, neg_b, B, c_mod, C, reuse_a, reuse_b)
    return __builtin_amdgcn_wmma_f32_16x16x4_f32(false, a, false, b, (short)0, c,
                                                 false, false);
}

// Async global->LDS 128-bit copy (VDST = LDS byte address, VADDR = 64-bit ptr).
__device__ __forceinline__ void async_g2l_b128(unsigned lds_byte, const float* g) {
    asm volatile("global_load_async_to_lds_b128 %0, %1, off"
                 :: "v"(lds_byte), "v"(g)
                 : "memory");
}

__global__ __launch_bounds__(384)
void win_attn_proj_kernel(const float* __restrict__ q,
                          const float* __restrict__ k,
                          const float* __restrict__ v,
                          const float* __restrict__ mask,
                          const float* __restrict__ rpe,
                          const float* __restrict__ pw,
                          const float* __restrict__ pb,
                          float* __restrict__ out) {
    extern __shared__ float lds[];
    float* sQ   = lds + LDS_Q;   // raw q (dead after stage B)
    float* sK   = lds + LDS_K;   // raw k (dead after stage B)
    float* sV   = lds + LDS_V;
    float* sS   = lds + LDS_S;   // scores/probs, per head 64 x SROW
    float* sInv = lds + LDS_I;   // per-row 1/sum(exp)
    float* sX   = lds;           // X_T[n][c], overlaps dead q/k region

    const int b    = blockIdx.x;
    const int tid  = threadIdx.x;
    const int wave = tid >> 5;
    const int lane = tid & 31;
    const int l16  = lane & 15;
    const int hi   = lane >> 4;      // lane-half: A/B K-offset +2, C/D M-offset +8

    // ---------------- Stage A: async-copy window into LDS --------------------
    // q/k/v are contiguous 48 KiB chunks per window; 9216 b128 transfers total,
    // 24 per thread, straight to LDS via the gfx1250 async path (no VGPR data).
    {
        const unsigned lbase =
            (unsigned)(unsigned long long)(lds_f*)lds;   // LDS byte addr of base
        const size_t base = (size_t)b * (CCH * NN);
        const float* qg = q + base;
        const float* kg = k + base;
        const float* vg = v + base;
        #pragma unroll
        for (int j = 0; j < 8; ++j) {
            const unsigned i = tid + j * 384;            // float4 index < 3072
            async_g2l_b128(lbase + (LDS_Q * 4u) + i * 16u, qg + i * 4);
            async_g2l_b128(lbase + (LDS_K * 4u) + i * 16u, kg + i * 4);
            async_g2l_b128(lbase + (LDS_V * 4u) + i * 16u, vg + i * 4);
        }
        asm volatile("s_wait_asynccnt 0x0" ::: "memory");
    }
    __syncthreads();

    // ---------------- Stage B: S = scale*(Q^T K) -----------------------------
    // Wave w owns head h=w>>1, rows n0..n0+31 (two 16-row tiles sharing the
    // same head): Q fragments cached once; per m-tile all 8 K fragments are
    // preloaded so the 16 WMMAs issue as one burst behind a single wait.
    const float scale = 0.17677669529663687f;            // 32^-0.5
    {
        const int h  = wave >> 1;
        const int n0 = (wave & 1) * 32;
        v2f aq0[8], aq1[8];
        #pragma unroll
        for (int kk = 0; kk < 8; ++kk) {                 // A[n][d] = q[d][n]
            const int dbase = h * DH + kk * 4 + hi * 2;
            aq0[kk].x = sQ[(dbase)     * NN + n0 + l16];
            aq0[kk].y = sQ[(dbase + 1) * NN + n0 + l16];
            aq1[kk].x = sQ[(dbase)     * NN + n0 + 16 + l16];
            aq1[kk].y = sQ[(dbase + 1) * NN + n0 + 16 + l16];
        }
        for (int mt = 0; mt < 4; ++mt) {
            const int m0 = mt * 16;
            v2f bk[8];
            #pragma unroll
            for (int kk = 0; kk < 8; ++kk) {             // B[d][m] = k[d][m]
                const int dbase = h * DH + kk * 4 + hi * 2;
                bk[kk].x = sK[(dbase)     * NN + m0 + l16];
                bk[kk].y = sK[(dbase + 1) * NN + m0 + l16];
            }
            v8f acc0 = {}, acc1 = {};
            #pragma unroll
            for (int kk = 0; kk < 8; ++kk) {             // 16-WMMA burst
                acc0 = wmma_f32(aq0[kk], bk[kk], acc0);
                acc1 = wmma_f32(aq1[kk], bk[kk], acc1);
            }
            float* s0 = sS + h * (NN * SROW) + n0 * SROW + m0;
            float* s1 = s0 + 16 * SROW;
            #pragma unroll
            for (int r = 0; r < 8; ++r) {                // C/D row r / r+8
                s0[(r + hi * 8) * SROW + l16] = acc0[r] * scale;
                s1[(r + hi * 8) * SROW + l16] = acc1[r] * scale;
            }
        }
    }
    __syncthreads();

    // ------- Stage C: row softmax, rpe+mask added with coalesced b128 --------
    // 384 rows / 384 threads; each thread owns one contiguous (h, n) row.
    {
        const int h = tid >> 6;
        const int n = tid & 63;
        float* row = sS + h * (NN * SROW) + n * SROW;
        const v4f* r4 = (const v4f*)(rpe  + ((size_t)h * NN + n) * NN);
        const v4f* m4 = (const v4f*)(mask + ((size_t)b * NN + n) * NN);
        float mx = -3.402823466e38f;
        #pragma unroll 4
        for (int i = 0; i < 16; ++i) {
            const v4f rv = r4[i];
            const v4f mv = __builtin_nontemporal_load(m4 + i); // streamed once
            const float t0 = row[4 * i + 0] + rv.x + mv.x;
            const float t1 = row[4 * i + 1] + rv.y + mv.y;
            const float t2 = row[4 * i + 2] + rv.z + mv.z;
            const float t3 = row[4 * i + 3] + rv.w + mv.w;
            row[4 * i + 0] = t0;
            row[4 * i + 1] = t1;
            row[4 * i + 2] = t2;
            row[4 * i + 3] = t3;
            mx = fmaxf(mx, fmaxf(fmaxf(t0, t1), fmaxf(t2, t3)));
        }
        float s = 0.f;
        #pragma unroll 8
        for (int i = 0; i < NN; ++i) {
            const float e = __expf(row[i] - mx);
            row[i] = e;                                  // unnormalized prob
            s += e;
        }
        sInv[tid] = 1.f / s;                             // applied in stage D
    }
    __syncthreads();

    // ------- Stage D: X_T[n][c] = (P * V^T)^T over dead q/k region -----------
    // Wave w owns (head h=w>>1, d-tile w&1); V fragments cached across the 4
    // n-tiles; per n-tile all 16 P fragments (b64) preload, then 16-WMMA burst;
    // softmax 1/sum is per-lane at store; X_T stores are two b128 per tile.
    {
        const int h   = wave >> 1;
        const int dd0 = h * DH + (wave & 1) * 16;        // absolute channel base
        const float* sP = sS + h * (NN * SROW);
        v2f av[16];
        #pragma unroll
        for (int kk = 0; kk < 16; ++kk)                  // A[dd][m] = v[dd][m]
            av[kk] = *(const v2f*)&sV[(dd0 + l16) * NN + kk * 4 + hi * 2];
        for (int nt = 0; nt < 4; ++nt) {
            const int n0 = nt * 16;
            v2f bp[16];
            #pragma unroll
            for (int kk = 0; kk < 16; ++kk)              // B[m][n] = P[n][m]
                bp[kk] = *(const v2f*)&sP[(n0 + l16) * SROW + kk * 4 + hi * 2];
            v8f acc = {};
            #pragma unroll
            for (int kk = 0; kk < 16; ++kk)              // 16-WMMA burst
                acc = wmma_f32(av[kk], bp[kk], acc);
            const float invn = sInv[h * NN + n0 + l16];  // row-n normalizer
            v4f lo, hi4;
            lo.x  = acc[0] * invn;  lo.y  = acc[1] * invn;
            lo.z  = acc[2] * invn;  lo.w  = acc[3] * invn;
            hi4.x = acc[4] * invn;  hi4.y = acc[5] * invn;
            hi4.z = acc[6] * invn;  hi4.w = acc[7] * invn;
            // lane holds 8 consecutive channels of row n -> two b128 stores
            float* px = sX + (n0 + l16) * XST + dd0 + hi * 8;
            *(v4f*)(px)     = lo;
            *(v4f*)(px + 4) = hi4;
        }
    }
    __syncthreads();

    // ---------------- Stage E: Y = W * X + b ---------------------------------
    // 12 o-tiles x 4 n-tiles; wave w owns o-tile w, all 4 n-accumulators so the
    // global W fragment (b64) is loaded once per K-step and reused 4x; X_T
    // makes each B fragment a single ds_load_b64. Unroll 4 -> clause batching.
    {
        const int o0 = wave * 16;
        v8f acc0 = {}, acc1 = {}, acc2 = {}, acc3 = {};
        #pragma unroll 4
        for (int kk = 0; kk < 48; ++kk) {                // K = c = 192
            const int c0 = kk * 4 + hi * 2;
            const v2f a  = *(const v2f*)&pw[(size_t)(o0 + l16) * CCH + c0];
            const v2f b0 = *(const v2f*)&sX[( 0 + l16) * XST + c0];
            const v2f b1 = *(const v2f*)&sX[(16 + l16) * XST + c0];
            const v2f b2 = *(const v2f*)&sX[(32 + l16) * XST + c0];
            const v2f b3 = *(const v2f*)&sX[(48 + l16) * XST + c0];
            acc0 = wmma_f32(a, b0, acc0);
            acc1 = wmma_f32(a, b1, acc1);
            acc2 = wmma_f32(a, b2, acc2);
            acc3 = wmma_f32(a, b3, acc3);
        }
        float* og = out + (size_t)b * (CCH * NN);
        #pragma unroll
        for (int r = 0; r < 8; ++r) {
            const int o = o0 + r + hi * 8;
            const float bias = pb[o];
            og[o * NN +  0 + l16] = acc0[r] + bias;
            og[o * NN + 16 + l16] = acc1[r] + bias;
            og[o * NN + 32 + l16] = acc2[r] + bias;
            og[o * NN + 48 + l16] = acc3[r] + bias;
        }
    }
}

extern "C" void kernel_launch(void* const* d_in, const int* in_sizes, int n_in,
                              void* d_out, int out_size, void* d_ws, size_t ws_size,
                              hipStream_t stream) {
    (void)in_sizes; (void)n_in; (void)out_size; (void)d_ws; (void)ws_size;
    const float* q    = (const float*)d_in[0];
    const float* k    = (const float*)d_in[1];
    const float* v    = (const float*)d_in[2];
    const float* mask = (const float*)d_in[3];
    const float* rpe  = (const float*)d_in[4];
    const float* pw   = (const float*)d_in[5];
    const float* pb   = (const float*)d_in[6];
    float* out        = (float*)d_out;

    const size_t lds_bytes = (size_t)LDS_FLOATS * sizeof(float);  // 250368 B
    win_attn_proj_kernel<<<NWIN, 384, lds_bytes, stream>>>(q, k, v, mask, rpe,
                                                           pw, pb, out);
}